// PolarRecurrent_57535381897479
// MI455X (gfx1250) — compile-verified
//
#include <hip/hip_runtime.h>
#include <math.h>

typedef __attribute__((ext_vector_type(2))) float v2f;
typedef __attribute__((ext_vector_type(8))) float v8f;

#define F_DT (1.0f / 240.0f)

// ---------------------------------------------------------------------------
// Precompute (tiny): per-(b,t) decay & EMA coefficients, per-(s,b,m) phase trig
//   coeff : [B][16][48]  row0=c_fast, row1=c_slow, row2=pick(t=47)*d47,
//                        row3=pick(t=46)*d46, rows4..15 = 0
//   bscal : [B][4]       d47, d46, Kf=(Σw_f-Σc_f), Ks=(Σw_s-Σc_s)
//   trig  : [S][B][M][4] cos47, sin47, cos45, sin45
// ---------------------------------------------------------------------------
__global__ void precompute_kernel(const float* __restrict__ q,
                                  const float* __restrict__ v_eff,
                                  const float* __restrict__ gamma,
                                  const float* __restrict__ rho,
                                  const float* __restrict__ lam_fix,
                                  const float* __restrict__ lam_sac,
                                  const float* __restrict__ kxy,
                                  float* __restrict__ coeff,
                                  float* __restrict__ bscal,
                                  float* __restrict__ trig) {
  const int B = 4, T = 48, M = 8;
  int tid = threadIdx.x;
  float lf = lam_fix[0], ls = lam_sac[0];

  // zero rows 2..15 of coeff cooperatively
  for (int i = tid; i < B * 16 * T; i += blockDim.x) {
    int row = (i / T) & 15;
    if (row >= 2) coeff[i] = 0.0f;
  }
  __syncthreads();

  if (tid < B) {
    int b = tid;
    float swf = 0.f, scf = 0.f, sws = 0.f, scs = 0.f, d47 = 0.f, d46 = 0.f;
    for (int t = 0; t < T; ++t) {
      float qv  = q[b * T + t];
      float lam = (1.0f - qv) * lf + qv * ls;
      float d   = expf(-lam * F_DT);
      float w_f = 0.26f * powf(0.74f, (float)(T - 1 - t));
      float w_s = 0.05f * powf(0.95f, (float)(T - 1 - t));
      float cf = w_f * d, cs = w_s * d;
      coeff[(b * 16 + 0) * T + t] = cf;
      coeff[(b * 16 + 1) * T + t] = cs;
      swf += w_f; scf += cf; sws += w_s; scs += cs;
      if (t == T - 1) d47 = d;
      if (t == T - 2) d46 = d;
    }
    coeff[(b * 16 + 2) * T + (T - 1)] = d47;
    coeff[(b * 16 + 3) * T + (T - 2)] = d46;
    bscal[b * 4 + 0] = d47;
    bscal[b * 4 + 1] = d46;
    bscal[b * 4 + 2] = swf - scf;
    bscal[b * 4 + 3] = sws - scs;
  }

  if (tid >= 64 && tid < 128) {
    int idx = tid - 64;
    int s = idx >> 5, b = (idx >> 3) & 3, m = idx & 7;
    float kx = kxy[(s * M + m) * 2 + 0];
    float ky = kxy[(s * M + m) * 2 + 1];
    for (int j = 0; j < 2; ++j) {
      int t = (j == 0) ? (T - 1) : (T - 3);
      float vx = v_eff[(b * T + t) * 2 + 0];
      float vy = v_eff[(b * T + t) * 2 + 1];
      float kv = kx * vx + ky * vy;
      float g = gamma[((b * T + t) * 2 + s) * M + m];
      float r = rho[((b * T + t) * 2 + s) * M + m];
      float dphi = g * kv * F_DT + r;
      trig[(((s * 4) + b) * M + m) * 4 + j * 2 + 0] = cosf(dphi);
      trig[(((s * 4) + b) * M + m) * 4 + j * 2 + 1] = sinf(dphi);
    }
  }
}

// ---------------------------------------------------------------------------
// Main level kernel. Block = 128 threads = 4 waves; each wave owns 32 pixels
// of one (b,m) plane. The T=48 reduction is done as 12 chained
// V_WMMA_F32_16X16X4_F32 ops per 16-pixel subtile:
//   A tile (16 pix x 4 t), B tile (4 t x 16 cols): col0=c_fast, col1=c_slow,
//   col2=last-pick, col3=prev-pick, cols 4..15 zero.
// D layout (lane L, vgpr r): row M = r + 8*(L>=16) = pixel, col N = L%16.
// Results transposed via LDS for fully coalesced channel stores.
// ---------------------------------------------------------------------------
__global__ void __launch_bounds__(128) level_kernel(
    const float* __restrict__ A,      // [B*M][T][HW]
    const float* __restrict__ U,      // [B*M][2][T][HW]
    const float* __restrict__ coeff,  // [B][16][48]
    const float* __restrict__ bscal,  // [B][4]
    const float* __restrict__ trig,   // [S][B][M][4]
    const float* __restrict__ a_bar,  // [S][M]
    float* __restrict__ out,          // [B][5*M][HW]
    int HW, int s) {
  const int T = 48, M = 8;
  int lane = threadIdx.x & 31;
  int wave = threadIdx.x >> 5;
  int N = lane & 15;      // A row = pixel-in-subtile, B col = reduction id
  int half = lane >> 4;   // K offset selector
  int bm = blockIdx.y;
  int b = bm >> 3, m = bm & 7;
  int wave_base = blockIdx.x * 128 + wave * 32;

  const float* Abm  = A + (size_t)bm * T * HW;
  const float* crow = coeff + (size_t)(b * 16 + N) * T;  // rows 4..15 are zero

  v8f D0 = {0.f, 0.f, 0.f, 0.f, 0.f, 0.f, 0.f, 0.f};
  v8f D1 = D0;
  int p0 = wave_base + N;        // subtile 0 pixel
  int p1 = wave_base + 16 + N;   // subtile 1 pixel

  for (int k0 = 0; k0 < T; k0 += 4) {
    int t0 = k0 + 2 * half;
    v2f bv;
    bv.x = crow[t0];
    bv.y = crow[t0 + 1];
    const float* row0 = Abm + (size_t)t0 * HW;
    const float* row1 = row0 + HW;
    v2f a0; a0.x = row0[p0]; a0.y = row1[p0];
    v2f a1; a1.x = row0[p1]; a1.y = row1[p1];
    D0 = __builtin_amdgcn_wmma_f32_16x16x4_f32(false, a0, false, bv,
                                               (short)0, D0, false, false);
    D1 = __builtin_amdgcn_wmma_f32_16x16x4_f32(false, a1, false, bv,
                                               (short)0, D1, false, false);
  }

  // LDS transpose: columns 0..3 hold {fast,slow,last,prev} sums per pixel
  __shared__ float xbuf[4][4][32];
  if (N < 4) {
#pragma unroll
    for (int r = 0; r < 8; ++r) {
      xbuf[wave][N][half * 8 + r]      = D0[r];
      xbuf[wave][N][16 + half * 8 + r] = D1[r];
    }
  }
  __syncthreads();

  float d47 = bscal[b * 4 + 0];
  float d46 = bscal[b * 4 + 1];
  float Kf  = bscal[b * 4 + 2];
  float Ks  = bscal[b * 4 + 3];
  float ab  = a_bar[s * M + m];
  int gp = wave_base + lane;

  float last  = xbuf[wave][2][lane] + ab * (1.0f - d47);
  float prevv = xbuf[wave][3][lane] + ab * (1.0f - d46);
  float fast  = xbuf[wave][0][lane] + ab * Kf;
  float slow  = xbuf[wave][1][lane] + ab * Ks;
  float deriv = last - prevv;

  // E_temp: only t = T-1 and T-3 slices of U are live
  const float* Ure = U + (size_t)bm * 2 * T * HW;
  const float* Uim = Ure + (size_t)T * HW;
  float ur1 = Ure[(size_t)(T - 1) * HW + gp];
  float ur3 = Ure[(size_t)(T - 3) * HW + gp];
  float ui1 = Uim[(size_t)(T - 1) * HW + gp];
  float ui3 = Uim[(size_t)(T - 3) * HW + gp];
  const float* tr = trig + (size_t)(((s * 4) + b) * M + m) * 4;
  float c1 = tr[0], s1 = tr[1], c3 = tr[2], s3 = tr[3];
  float re1 = c1 * ur1 - s1 * ui1;
  float im1 = s1 * ur1 + c1 * ui1;
  float re3 = c3 * ur3 - s3 * ui3;
  float im3 = s3 * ur3 + c3 * ui3;
  float qre = 0.5f * (re1 - re3);
  float qim = 0.5f * (im1 - im3);
  float er2 = fmaxf(re1 * re1 + qre * qre, 1e-6f);
  float ei2 = fmaxf(im1 * im1 + qim * qim, 1e-6f);
  float E = sqrtf(fmaxf(er2 + ei2, 1e-6f));

  // out channel = b*5M + cat*M + m ; fully coalesced 32-lane stores
  float* ob = out + ((size_t)b * 5 * M + m) * HW + gp;
  size_t cstr = (size_t)M * HW;
  ob[0]        = last;
  ob[cstr]     = fast;
  ob[2 * cstr] = slow;
  ob[3 * cstr] = deriv;
  ob[4 * cstr] = E;
}

extern "C" void kernel_launch(void* const* d_in, const int* in_sizes, int n_in,
                              void* d_out, int out_size, void* d_ws, size_t ws_size,
                              hipStream_t stream) {
  (void)in_sizes; (void)n_in; (void)out_size; (void)ws_size;
  const float* A0      = (const float*)d_in[0];
  const float* U0      = (const float*)d_in[1];
  const float* A1      = (const float*)d_in[2];
  const float* U1      = (const float*)d_in[3];
  const float* q       = (const float*)d_in[4];
  const float* v_eff   = (const float*)d_in[5];
  const float* gamma   = (const float*)d_in[6];
  const float* rho     = (const float*)d_in[7];
  const float* lam_fix = (const float*)d_in[8];
  const float* lam_sac = (const float*)d_in[9];
  const float* a_bar   = (const float*)d_in[10];
  const float* kxy     = (const float*)d_in[11];
  float* out = (float*)d_out;

  float* coeff = (float*)d_ws;            // 4*16*48 = 3072 floats
  float* bscal = coeff + 4 * 16 * 48;     // 16 floats
  float* trig  = bscal + 16;              // 256 floats

  precompute_kernel<<<1, 256, 0, stream>>>(q, v_eff, gamma, rho, lam_fix,
                                           lam_sac, kxy, coeff, bscal, trig);

  const int HW0 = 128 * 128, HW1 = 64 * 64;
  dim3 grid0(HW0 / 128, 32);
  dim3 grid1(HW1 / 128, 32);
  level_kernel<<<grid0, 128, 0, stream>>>(A0, U0, coeff, bscal, trig, a_bar,
                                          out, HW0, 0);
  level_kernel<<<grid1, 128, 0, stream>>>(A1, U1, coeff, bscal, trig, a_bar,
                                          out + (size_t)4 * 40 * HW0, HW1, 1);
}